// CustomBiMamba_36627481100484
// MI455X (gfx1250) — compile-verified
//
#include <hip/hip_runtime.h>
#include <hip/hip_bf16.h>

#define D_MODEL 128
#define D_INNER 256
#define D_STATE 16
#define D_CONV  4
#define DT_RANK 8
#define BB 4
#define TT 24
#define NSEQ 207
#define EPSV 1e-5f

typedef __attribute__((ext_vector_type(2))) float v2f;
typedef __attribute__((ext_vector_type(8))) float v8f;

__device__ __forceinline__ float wave_sum(float v) {
#pragma unroll
  for (int off = 16; off >= 1; off >>= 1) v += __shfl_xor(v, off, 32);
  return v;
}

__device__ __forceinline__ float silu_(float x) { return x / (1.f + __expf(-x)); }
__device__ __forceinline__ float softplus_(float x) {
  return (x > 20.f) ? x : log1pf(__expf(x));
}

// D(16x16) += A(16xK from LDS, row stride lda) * B(KxN), B[k][n] = W[n*ldw + k].
// Lanes with n >= nmax contribute zero (mask by multiply; no EXEC divergence).
__device__ __forceinline__ v8f wmma_mm(const float* Ash, int lda, int m0,
                                       const float* __restrict__ W, int ldw,
                                       int n0, int nmax, int K) {
  const int lane = threadIdx.x & 31;
  const int half = lane >> 4;      // 0: K=0,1   1: K=2,3
  const int l15  = lane & 15;
  const int kk   = half * 2;
  const int n    = n0 + l15;
  const float bmask = (n < nmax) ? 1.f : 0.f;
  const int   nc    = (n < nmax) ? n : (nmax - 1);
  const float* arow = Ash + (m0 + l15) * lda + kk;
  const float* brow = W + (size_t)nc * ldw + kk;
  v8f acc = {};
  for (int k = 0; k < K; k += 4) {
    v2f a, b;
    a.x = arow[k];  a.y = arow[k + 1];
    b.x = brow[k] * bmask;  b.y = brow[k + 1] * bmask;
    acc = __builtin_amdgcn_wmma_f32_16x16x4_f32(false, a, false, b,
                                                (short)0, acc, false, false);
  }
  return acc;
}

// Store 16x16 f32 C/D fragment: VGPR r -> M = m0 + r + 8*half, N = n0 + (lane&15).
__device__ __forceinline__ void wmma_st(float* O, int ldo, int m0, int mmax,
                                        int n0, v8f acc) {
  const int lane = threadIdx.x & 31;
  const int half = lane >> 4;
  const int l15  = lane & 15;
#pragma unroll
  for (int r = 0; r < 8; ++r) {
    int m = m0 + half * 8 + r;
    if (m < mmax) O[m * ldo + n0 + l15] = acc[r];
  }
}

__global__ __launch_bounds__(256)
void bimamba_fused(const float* __restrict__ x,
                   const float* __restrict__ ln1_g, const float* __restrict__ ln1_b,
                   const float* __restrict__ in_proj_w,
                   const float* __restrict__ conv_w,  const float* __restrict__ conv_b,
                   const float* __restrict__ xproj_w,
                   const float* __restrict__ dtproj_w, const float* __restrict__ dtproj_b,
                   const float* __restrict__ A_log,   const float* __restrict__ Dp,
                   const float* __restrict__ conv_w_b, const float* __restrict__ conv_b_b,
                   const float* __restrict__ xproj_w_b,
                   const float* __restrict__ dtproj_w_b, const float* __restrict__ dtproj_b_b,
                   const float* __restrict__ A_b_log, const float* __restrict__ Dp_b,
                   const float* __restrict__ out_proj_w,
                   const float* __restrict__ ln2_g, const float* __restrict__ ln2_b,
                   float* __restrict__ out) {
  __shared__ float u_sh[TT][D_MODEL];        // 12 KB, residual (live to end)
  __shared__ float hA[32][260];              // 32.5 KB, padded A-operand staging
  __shared__ float xz_sh[TT][2 * D_INNER];   // 48 KB: xx | z, later out_proj result
  __shared__ float xdbl_sh[2][TT][48];       // 9 KB: dt_in(8) | B(16) | C(16) per branch

  const int tid  = threadIdx.x;
  const int lane = tid & 31;
  const int wave = tid >> 5;
  const int seq  = blockIdx.x;
  const int bidx = seq / NSEQ;
  const int nidx = seq % NSEQ;

  // ---- load u (B,T,N,C layout) ----
  for (int i = tid; i < TT * D_MODEL; i += 256) {
    int t = i >> 7, c = i & 127;
    u_sh[t][c] = x[(((size_t)bidx * TT + t) * NSEQ + nidx) * D_MODEL + c];
  }
  // zero pad rows 24..31 of A staging (once; later stages only write rows 0..23)
  for (int i = tid; i < 8 * 260; i += 256) hA[24 + i / 260][i % 260] = 0.f;
  __syncthreads();

  // ---- ln1 -> hA[0..23][0..127] (one wave per token) ----
  for (int t = wave; t < TT; t += 8) {
    float s = 0.f;
#pragma unroll
    for (int j = 0; j < 4; ++j) s += u_sh[t][lane * 4 + j];
    float mean = wave_sum(s) * (1.f / 128.f);
    float v = 0.f;
#pragma unroll
    for (int j = 0; j < 4; ++j) { float d = u_sh[t][lane * 4 + j] - mean; v += d * d; }
    float rstd = rsqrtf(wave_sum(v) * (1.f / 128.f) + EPSV);
#pragma unroll
    for (int j = 0; j < 4; ++j) {
      int c = lane * 4 + j;
      hA[t][c] = (u_sh[t][c] - mean) * rstd * ln1_g[c] + ln1_b[c];
    }
  }
  __syncthreads();

  // ---- in_proj: xz(24x512) = h(24x128) * W^T, WMMA f32 16x16x4, 64 tiles ----
  for (int tile = wave; tile < 64; tile += 8) {
    int mt = tile & 1, nt = tile >> 1;
    v8f acc = wmma_mm(&hA[0][0], 260, mt * 16, in_proj_w, D_MODEL, nt * 16, 512, D_MODEL);
    wmma_st(&xz_sh[0][0], 2 * D_INNER, mt * 16, TT, nt * 16, acc);
  }
  __syncthreads();

  // ---- per-branch: conv+silu -> hA, xproj (WMMA) -> xdbl ----
  for (int br = 0; br < 2; ++br) {
    const float* cw = br ? conv_w_b : conv_w;
    const float* cb = br ? conv_b_b : conv_b;
    const float* xw = br ? xproj_w_b : xproj_w;
    {
      const int d = tid;  // channel per thread
      const float w0 = cw[d * 4 + 0], w1 = cw[d * 4 + 1];
      const float w2 = cw[d * 4 + 2], w3 = cw[d * 4 + 3];
      const float cbv = cb[d];
      for (int t = 0; t < TT; ++t) {
        float acc = cbv;
        // causal conv over (possibly time-reversed) xx
#pragma unroll
        for (int j = 0; j < 4; ++j) {
          int ts = t - 3 + j;
          if (ts >= 0) {
            int tt = br ? (TT - 1 - ts) : ts;
            float wj = (j == 0) ? w0 : (j == 1) ? w1 : (j == 2) ? w2 : w3;
            acc += wj * xz_sh[tt][d];
          }
        }
        hA[t][d] = silu_(acc);
      }
    }
    __syncthreads();
    // xproj: (24x256)@(256x40) -> xdbl, pad N to 48 (lanes >=40 masked)
    for (int tile = wave; tile < 6; tile += 8) {
      int mt = tile & 1, nt = tile >> 1;
      v8f acc = wmma_mm(&hA[0][0], 260, mt * 16, xw, D_INNER, nt * 16,
                        DT_RANK + 2 * D_STATE, D_INNER);
      wmma_st(&xdbl_sh[br][0][0], 48, mt * 16, TT, nt * 16, acc);
    }
    __syncthreads();
  }

  // ---- selective scans (channel d = tid), write gated y into hA rows 0..23 ----
  {
    const int d = tid;
    for (int br = 0; br < 2; ++br) {
      const float* cw  = br ? conv_w_b : conv_w;
      const float* cb  = br ? conv_b_b : conv_b;
      const float* dwp = br ? dtproj_w_b : dtproj_w;
      const float* dbp = br ? dtproj_b_b : dtproj_b;
      const float* Alg = br ? A_b_log : A_log;
      const float* Dpp = br ? Dp_b : Dp;

      float Arow[D_STATE];
#pragma unroll
      for (int sx = 0; sx < D_STATE; ++sx) Arow[sx] = -__expf(Alg[d * D_STATE + sx]);
      float dwr[DT_RANK];
#pragma unroll
      for (int r = 0; r < DT_RANK; ++r) dwr[r] = dwp[d * DT_RANK + r];
      const float dbv = dbp[d], Dpd = Dpp[d];
      const float w0 = cw[d * 4 + 0], w1 = cw[d * 4 + 1];
      const float w2 = cw[d * 4 + 2], w3 = cw[d * 4 + 3];
      const float cbv = cb[d];

      float hs[D_STATE];
#pragma unroll
      for (int sx = 0; sx < D_STATE; ++sx) hs[sx] = 0.f;

      for (int t = 0; t < TT; ++t) {
        // recompute conv+silu (4 MACs, avoids a 48 KB xc buffer)
        float u = cbv;
#pragma unroll
        for (int j = 0; j < 4; ++j) {
          int ts = t - 3 + j;
          if (ts >= 0) {
            int tt = br ? (TT - 1 - ts) : ts;
            float wj = (j == 0) ? w0 : (j == 1) ? w1 : (j == 2) ? w2 : w3;
            u += wj * xz_sh[tt][d];
          }
        }
        u = silu_(u);
        float p = dbv;
#pragma unroll
        for (int r = 0; r < DT_RANK; ++r) p += xdbl_sh[br][t][r] * dwr[r];
        float dt = softplus_(p);
        float du = dt * u;
        float yacc = 0.f;
#pragma unroll
        for (int sx = 0; sx < D_STATE; ++sx) {
          hs[sx] = hs[sx] * __expf(dt * Arow[sx]) + du * xdbl_sh[br][t][8 + sx];
          yacc += hs[sx] * xdbl_sh[br][t][24 + sx];
        }
        int to = br ? (TT - 1 - t) : t;        // original time index
        float zv = xz_sh[to][D_INNER + d];     // gate in original time
        float yv = (yacc + Dpd * u) * silu_(zv);
        if (br == 0) hA[to][d] = yv; else hA[to][d] += yv;
      }
    }
  }
  __syncthreads();

  // ---- out_proj: o(24x128) = y(24x256) * W^T, 16 tiles; store into xz_sh[.][0..127] ----
  for (int tile = wave; tile < 16; tile += 8) {
    int mt = tile & 1, nt = tile >> 1;
    v8f acc = wmma_mm(&hA[0][0], 260, mt * 16, out_proj_w, D_INNER, nt * 16,
                      D_MODEL, D_INNER);
    wmma_st(&xz_sh[0][0], 2 * D_INNER, mt * 16, TT, nt * 16, acc);
  }
  __syncthreads();

  // ---- ln2 + residual -> out (B,T,N,C) ----
  for (int t = wave; t < TT; t += 8) {
    float s = 0.f;
#pragma unroll
    for (int j = 0; j < 4; ++j) s += xz_sh[t][lane * 4 + j];
    float mean = wave_sum(s) * (1.f / 128.f);
    float v = 0.f;
#pragma unroll
    for (int j = 0; j < 4; ++j) { float d = xz_sh[t][lane * 4 + j] - mean; v += d * d; }
    float rstd = rsqrtf(wave_sum(v) * (1.f / 128.f) + EPSV);
#pragma unroll
    for (int j = 0; j < 4; ++j) {
      int c = lane * 4 + j;
      float o = (xz_sh[t][c] - mean) * rstd * ln2_g[c] + ln2_b[c] + u_sh[t][c];
      out[(((size_t)bidx * TT + t) * NSEQ + nidx) * D_MODEL + c] = o;
    }
  }
}

extern "C" void kernel_launch(void* const* d_in, const int* in_sizes, int n_in,
                              void* d_out, int out_size, void* d_ws, size_t ws_size,
                              hipStream_t stream) {
  (void)in_sizes; (void)n_in; (void)d_ws; (void)ws_size; (void)out_size;
  const float* p[21];
  for (int i = 0; i < 21; ++i) p[i] = (const float*)d_in[i];
  dim3 grid(BB * NSEQ), block(256);
  bimamba_fused<<<grid, block, 0, stream>>>(
      p[0], p[1], p[2], p[3], p[4], p[5], p[6], p[7], p[8], p[9], p[10],
      p[11], p[12], p[13], p[14], p[15], p[16], p[17], p[18], p[19], p[20],
      (float*)d_out);
}